// MLSTMCell_83777632076208
// MI455X (gfx1250) — compile-verified
//
#include <hip/hip_runtime.h>
#include <hip/hip_bf16.h>
#include <math.h>

typedef __attribute__((ext_vector_type(2))) float v2f;
typedef __attribute__((ext_vector_type(4))) float v4f;
typedef __attribute__((ext_vector_type(8))) float v8f;

#define HIDDEN 1024
#define HEADS  8
#define HEAD   128
#define BATCH  512
#define K3     (3 * HIDDEN)

// ---------------------------------------------------------------------------
// Kernel 1: gate pre-activation GEMM via V_WMMA_F32_16X16X4_F32.
//   preact[row][n], n in [0,16): n<8 -> igate_preact, n>=8 -> fgate_preact
//   A = qkv tile (16 rows x 4 K), B = [Wi;Wf]^T tile (4 K x 16 outputs).
//   Grid: 32 blocks (16 batch rows each), 8 waves split K=3072 into 384-chunks.
// ---------------------------------------------------------------------------
__global__ __launch_bounds__(256) void gate_gemm_kernel(
    const float* __restrict__ q, const float* __restrict__ k,
    const float* __restrict__ v,
    const float* __restrict__ Wi, const float* __restrict__ Wf,
    const float* __restrict__ bi, const float* __restrict__ bf,
    float* __restrict__ preact)
{
    __shared__ float red[8 * 256];

    const int tid  = threadIdx.x;
    const int wave = tid >> 5;
    const int lane = tid & 31;
    const int tile = blockIdx.x;              // 16 batch rows per tile

    // A fragment addressing (32-bit A 16x4): lane<16 -> K=k0,k0+1 ; lane>=16 -> K=k0+2,k0+3
    const int mrow = tile * 16 + (lane & 15);
    const int kh   = (lane >> 4) << 1;        // 0 or 2
    // B fragment: same K split; N = lane&15
    const int n    = lane & 15;
    const float* wrow = (n < 8) ? (Wi + (size_t)n * K3) : (Wf + (size_t)(n - 8) * K3);

    v8f c = {};

    const int kbeg = wave * 384;
    const int kend = kbeg + 384;
    for (int k0 = kbeg; k0 < kend; k0 += 4) {
        // qkv concat: segment never straddles a 1024 boundary (k0 % 4 == 0)
        const int seg = k0 >> 10;
        const int off = k0 & 1023;
        const float* src = (seg == 0) ? q : (seg == 1) ? k : v;

        v2f a, b;
        const float* ap = src + (size_t)mrow * HIDDEN + off + kh;
        a.x = ap[0]; a.y = ap[1];
        const float* bp = wrow + k0 + kh;
        b.x = bp[0]; b.y = bp[1];

        // 8 args: (neg_a, A, neg_b, B, c_mod, C, reuse_a, reuse_b)
        c = __builtin_amdgcn_wmma_f32_16x16x4_f32(false, a, false, b,
                                                  (short)0, c, false, false);
    }

    // per-wave partial tiles -> LDS -> cross-wave reduction
#pragma unroll
    for (int r = 0; r < 8; ++r)
        red[wave * 256 + r * 32 + lane] = c[r];
    __syncthreads();

    float s = 0.f;
#pragma unroll
    for (int w = 0; w < 8; ++w) s += red[w * 256 + tid];

    // C layout: VGPR r, lane<16: M=r,N=lane ; lane>=16: M=r+8,N=lane-16
    const int r2 = tid >> 5;
    const int l2 = tid & 31;
    const int M  = r2 + ((l2 & 16) ? 8 : 0);
    const int N  = l2 & 15;
    const int row = tile * 16 + M;
    s += (N < 8) ? bi[N] : bf[N - 8];
    preact[row * 16 + N] = s;
}

// ---------------------------------------------------------------------------
// Kernel 2: per-(b,head) mLSTM state update (bandwidth bound, ~537 MB stream).
//   One 256-thread block per (b,head). Non-temporal cell traffic (working set
//   537 MB > 192 MB L2). float4 coalesced access: thread t handles cols
//   4*(t%32)..4*(t%32)+3, rows (t/32)+8j.
// ---------------------------------------------------------------------------
__global__ __launch_bounds__(256) void mlstm_update_kernel(
    const float* __restrict__ q, const float* __restrict__ k,
    const float* __restrict__ v,
    const float* __restrict__ cell_state, const float* __restrict__ norm_state,
    const float* __restrict__ max_state,
    const float* __restrict__ gn_gamma, const float* __restrict__ gn_beta,
    const float* __restrict__ preact,
    float* __restrict__ out_y,      // (B, 1024)
    float* __restrict__ out_cell,   // (B, H, 128, 128)
    float* __restrict__ out_norm,   // (B, H, 128)
    float* __restrict__ out_max)    // (B, H)
{
    const int bh  = blockIdx.x;     // 0..4095
    const int b   = bh >> 3;
    const int h   = bh & 7;
    const int tid = threadIdx.x;

    __shared__ float sq[HEAD], skh[HEAD], sv[HEAD], snum[HEAD];
    __shared__ float s_qn, s_sum, s_sum2;

    // gates (recomputed redundantly per thread; scalar work is negligible)
    const float pre_i = preact[b * 16 + h];
    const float pre_f = preact[b * 16 + 8 + h];
    const float m_old = max_state[bh];
    const float log_f = fminf(pre_f, 0.f) - log1pf(expf(-fabsf(pre_f)));
    const float m_new = fmaxf(pre_i, m_old + log_f);
    const float i_g   = expf(pre_i - m_new);
    const float f_g   = expf(log_f + m_old - m_new);

    if (tid == 0) {
        out_max[bh] = m_new;
        s_qn = 0.f; s_sum = 0.f; s_sum2 = 0.f;
    }
    if (tid < HEAD) {
        const size_t base = (size_t)b * HIDDEN + (size_t)h * HEAD + tid;
        sq[tid]  = q[base];
        skh[tid] = k[base] * 0.08838834764831845f;   // 1/sqrt(128)
        sv[tid]  = v[base];
        snum[tid] = 0.f;
    }
    __syncthreads();

    // norm state update + qn_dot
    if (tid < HEAD) {
        const float nn = f_g * norm_state[(size_t)bh * HEAD + tid] + i_g * skh[tid];
        out_norm[(size_t)bh * HEAD + tid] = nn;
        atomicAdd(&s_qn, sq[tid] * nn);
    }

    // cell update + numerator accumulation (streaming, non-temporal)
    const int cg = tid & 31;
    const int r0 = tid >> 5;
    const float* cin  = cell_state + (size_t)bh * HEAD * HEAD;
    float*       cout = out_cell   + (size_t)bh * HEAD * HEAD;
    const float v0 = sv[4 * cg + 0], v1 = sv[4 * cg + 1];
    const float v2 = sv[4 * cg + 2], v3 = sv[4 * cg + 3];

    float acc0 = 0.f, acc1 = 0.f, acc2 = 0.f, acc3 = 0.f;
#pragma unroll 4
    for (int r = r0; r < HEAD; r += 8) {
        const float qr = sq[r];
        const float ik = i_g * skh[r];
        const v4f cold =
            __builtin_nontemporal_load((const v4f*)(cin + (size_t)r * HEAD + 4 * cg));
        v4f cnew;
        cnew.x = f_g * cold.x + ik * v0;
        cnew.y = f_g * cold.y + ik * v1;
        cnew.z = f_g * cold.z + ik * v2;
        cnew.w = f_g * cold.w + ik * v3;
        __builtin_nontemporal_store(cnew, (v4f*)(cout + (size_t)r * HEAD + 4 * cg));
        acc0 += qr * cnew.x;
        acc1 += qr * cnew.y;
        acc2 += qr * cnew.z;
        acc3 += qr * cnew.w;
    }
    atomicAdd(&snum[4 * cg + 0], acc0);
    atomicAdd(&snum[4 * cg + 1], acc1);
    atomicAdd(&snum[4 * cg + 2], acc2);
    atomicAdd(&snum[4 * cg + 3], acc3);
    __syncthreads();

    // output + GroupNorm statistics
    float o = 0.f;
    if (tid < HEAD) {
        const float denom = fmaxf(fabsf(s_qn), expf(-m_new)) + 1e-6f;
        o = snum[tid] / denom;
        atomicAdd(&s_sum, o);
        atomicAdd(&s_sum2, o * o);
    }
    __syncthreads();

    if (tid < HEAD) {
        const float mu   = s_sum * (1.f / HEAD);
        const float var  = s_sum2 * (1.f / HEAD) - mu * mu;
        const float rstd = rsqrtf(var + 1e-5f);
        const int ch = h * HEAD + tid;
        out_y[(size_t)b * HIDDEN + ch] = (o - mu) * rstd * gn_gamma[ch] + gn_beta[ch];
    }
}

// ---------------------------------------------------------------------------
extern "C" void kernel_launch(void* const* d_in, const int* in_sizes, int n_in,
                              void* d_out, int out_size, void* d_ws, size_t ws_size,
                              hipStream_t stream)
{
    const float* q     = (const float*)d_in[0];
    const float* k     = (const float*)d_in[1];
    const float* v     = (const float*)d_in[2];
    const float* cell  = (const float*)d_in[3];
    const float* norm  = (const float*)d_in[4];
    const float* maxs  = (const float*)d_in[5];
    const float* Wi    = (const float*)d_in[6];
    const float* bi    = (const float*)d_in[7];
    const float* Wf    = (const float*)d_in[8];
    const float* bf    = (const float*)d_in[9];
    const float* gamma = (const float*)d_in[10];
    const float* beta  = (const float*)d_in[11];

    float* preact = (float*)d_ws;                 // 512*16 floats = 32 KB

    // outputs concatenated flat in return order
    float* out_y    = (float*)d_out;
    float* out_cell = out_y    + (size_t)BATCH * HIDDEN;
    float* out_norm = out_cell + (size_t)BATCH * HEADS * HEAD * HEAD;
    float* out_max  = out_norm + (size_t)BATCH * HEADS * HEAD;

    gate_gemm_kernel<<<BATCH / 16, 256, 0, stream>>>(q, k, v, Wi, Wf, bi, bf, preact);
    mlstm_update_kernel<<<BATCH * HEADS, 256, 0, stream>>>(
        q, k, v, cell, norm, maxs, gamma, beta, preact,
        out_y, out_cell, out_norm, out_max);
}